// GCN_BJORK_11493332484238
// MI455X (gfx1250) — compile-verified
//
#include <hip/hip_runtime.h>
#include <math.h>

typedef __attribute__((ext_vector_type(2))) float v2f;
typedef __attribute__((ext_vector_type(8))) float v8f;

#define DIN 128
#define DOUT 64

// ---------------------------------------------------------------------------
// Kernel 1: zero the output accumulator and precompute per-node tables
//   de2[i] = safe_pow(d, e2), de3[i] = safe_pow(d, e3)
//   gso2[i] = m1*safe_pow(d,e1) + m2*de2*de3 + m3
// ---------------------------------------------------------------------------
__global__ __launch_bounds__(256) void precompute_kernel(
    const float* __restrict__ diags, float* __restrict__ out,
    float* __restrict__ de2, float* __restrict__ de3, float* __restrict__ gso2,
    const float* m1, const float* m2, const float* m3,
    const float* e1, const float* e2, const float* e3,
    int N, int total) {
  int i = blockIdx.x * 256 + threadIdx.x;
  if (i < total) out[i] = 0.0f;            // zero atomic accumulator (d_out)
  if (i < N) {
    float d = diags[i];
    float p1 = powf(d, e1[0]); if (isinf(p1)) p1 = 0.0f;
    float p2 = powf(d, e2[0]); if (isinf(p2)) p2 = 0.0f;
    float p3 = powf(d, e3[0]); if (isinf(p3)) p3 = 0.0f;
    de2[i] = p2;
    de3[i] = p3;
    gso2[i] = m1[0] * p1 + m2[0] * p2 * p3 + m3[0];
  }
}

// ---------------------------------------------------------------------------
// Kernel 2: Bjorck-Bowie orthogonalization of W (64x128), single workgroup.
//   sigma_max via power iteration on G = W W^T (64x64), then 5 iterations of
//   w = 1.5 w - 0.5 (w w^T) w, all LDS-resident (~48.5 KB)
// ---------------------------------------------------------------------------
__global__ __launch_bounds__(256) void bjorck_kernel(
    const float* __restrict__ W, float* __restrict__ Wo) {
  __shared__ float Ws[DOUT * DIN];     // 32 KB
  __shared__ float G[DOUT * DOUT];     // 16 KB
  __shared__ float vbuf[DOUT];
  __shared__ float ubuf[DOUT];
  __shared__ float red;

  const int tid = threadIdx.x;
  for (int i = tid; i < DOUT * DIN; i += 256) Ws[i] = W[i];
  __syncthreads();

  // G = Ws * Ws^T
  for (int idx = tid; idx < DOUT * DOUT; idx += 256) {
    int i = idx >> 6, j = idx & 63;
    float s = 0.0f;
    for (int k = 0; k < DIN; ++k) s += Ws[i * DIN + k] * Ws[j * DIN + k];
    G[idx] = s;
  }
  __syncthreads();

  if (tid < DOUT) vbuf[tid] = 1.0f;
  __syncthreads();

  // power iteration: lambda_max(G) = sigma_max(W)^2
  for (int it = 0; it < 96; ++it) {
    if (tid < DOUT) {
      float s = 0.0f;
      for (int j = 0; j < DOUT; ++j) s += G[tid * DOUT + j] * vbuf[j];
      ubuf[tid] = s;
    }
    __syncthreads();
    if (tid == 0) {
      float s = 0.0f;
      for (int j = 0; j < DOUT; ++j) s += ubuf[j] * ubuf[j];
      red = sqrtf(s);
    }
    __syncthreads();
    if (tid < DOUT) vbuf[tid] = ubuf[tid] / red;
    __syncthreads();
  }
  const float sigma = sqrtf(red);
  const float scale = 1.0f / (sigma * 1.1f);
  for (int i = tid; i < DOUT * DIN; i += 256) Ws[i] *= scale;
  __syncthreads();

  // 5 Bjorck iterations: w = 1.5 w - 0.5 (w w^T) w
  for (int iter = 0; iter < 5; ++iter) {
    for (int idx = tid; idx < DOUT * DOUT; idx += 256) {
      int i = idx >> 6, j = idx & 63;
      float s = 0.0f;
      for (int k = 0; k < DIN; ++k) s += Ws[i * DIN + k] * Ws[j * DIN + k];
      G[idx] = s;
    }
    __syncthreads();
    float nw[32];                      // each thread owns 32 contiguous elems
    for (int t = 0; t < 32; ++t) {
      int idx = tid * 32 + t;
      int i = idx >> 7, k = idx & 127;
      float s = 0.0f;
      for (int j = 0; j < DOUT; ++j) s += G[i * DOUT + j] * Ws[j * DIN + k];
      nw[t] = 1.5f * Ws[idx] - 0.5f * s;
    }
    __syncthreads();
    for (int t = 0; t < 32; ++t) Ws[tid * 32 + t] = nw[t];
    __syncthreads();
  }
  for (int i = tid; i < DOUT * DIN; i += 256) Wo[i] = Ws[i];
}

// ---------------------------------------------------------------------------
// Kernel 3: xw = x @ Wo^T via V_WMMA_F32_16X16X4_F32 (full f32 precision;
// memory-bound so f32 WMMA already saturates HBM). One wave computes a 16x64
// row-tile with 4 f32 accumulators over K=128 in steps of 4. Wo staged in LDS
// with stride 130 to break the 16-way bank conflict of a 512-byte lane stride.
//   A 16x4 layout: VGPR0 = K(0|2), VGPR1 = K(1|3); lanes 16-31 carry K+2.
//   B  4x16 layout mirrored: lanes = N, VGPRs/halves = K.
// Full tiles take the unguarded coalesced store path (N % 16 == 0 here).
// ---------------------------------------------------------------------------
__global__ __launch_bounds__(256) void gemm_xw_kernel(
    const float* __restrict__ x, const float* __restrict__ Wo,
    float* __restrict__ xw, int nRowTiles, int nRows) {
  __shared__ float Wl[DOUT * 130];
  const int tid = threadIdx.x;
  for (int idx = tid; idx < DOUT * DIN; idx += 256) {
    int n = idx >> 7, k = idx & 127;
    Wl[n * 130 + k] = Wo[idx];
  }
  __syncthreads();

  const int wave = tid >> 5;
  const int lane = tid & 31;
  const int half = lane >> 4;     // 0: K+0/1, 1: K+2/3
  const int lm   = lane & 15;     // M index (A) / N index (B)
  const int rt = blockIdx.x * 8 + wave;
  if (rt >= nRowTiles) return;

  const int row0 = rt * 16;
  int arow = row0 + lm;
  if (arow >= nRows) arow = nRows - 1;        // only hit in a ragged last tile
  const float* __restrict__ xr = x + (size_t)arow * DIN;

  v8f acc[4];
  #pragma unroll
  for (int c = 0; c < 4; ++c) acc[c] = (v8f){0.f,0.f,0.f,0.f,0.f,0.f,0.f,0.f};

  #pragma unroll 4
  for (int k0 = 0; k0 < DIN; k0 += 4) {
    v2f a = *(const v2f*)(xr + k0 + 2 * half);
    #pragma unroll
    for (int c = 0; c < 4; ++c) {
      const int n = c * 16 + lm;
      v2f bf = *(const v2f*)(&Wl[n * 130 + k0 + 2 * half]);
      acc[c] = __builtin_amdgcn_wmma_f32_16x16x4_f32(
          /*neg_a=*/false, a, /*neg_b=*/false, bf,
          /*c_mod=*/(short)0, acc[c], /*reuse_a=*/false, /*reuse_b=*/false);
    }
  }

  // D layout: VGPR r -> M = r + 8*half, N = lm
  float* __restrict__ outp = xw + (size_t)row0 * DOUT;
  if (row0 + 16 <= nRows) {
    // full tile: unguarded, coalesced b32 stores
    #pragma unroll
    for (int c = 0; c < 4; ++c) {
      #pragma unroll
      for (int r = 0; r < 8; ++r) {
        const int m = r + 8 * half;
        outp[(size_t)m * DOUT + c * 16 + lm] = acc[c][r];
      }
    }
  } else {
    #pragma unroll
    for (int c = 0; c < 4; ++c) {
      #pragma unroll
      for (int r = 0; r < 8; ++r) {
        const int m = r + 8 * half;
        if (row0 + m < nRows)
          outp[(size_t)m * DOUT + c * 16 + lm] = acc[c][r];
      }
    }
  }
}

// ---------------------------------------------------------------------------
// Kernel 4: edge scatter h[col] += m2*de2[row]*de3[col] * xw[row]
// One wave32 per edge, 2 features per lane. Edge id is forced into an SGPR so
// row/col/de2/de3 become scalar loads (KMcnt path) instead of 32-lane
// broadcast VMEM. Atomics + gathers all hit the 192 MB L2 (xw + h ~= 51 MB).
// ---------------------------------------------------------------------------
__global__ __launch_bounds__(256) void edge_scatter_kernel(
    const int* __restrict__ ei, const float* __restrict__ de2,
    const float* __restrict__ de3, const float* __restrict__ xw,
    const float* m2p, float* __restrict__ h, int E) {
  const int gid = blockIdx.x * 256 + threadIdx.x;
  const int lane = threadIdx.x & 31;
  const int e = __builtin_amdgcn_readfirstlane(gid >> 5);  // wave-uniform
  if (e >= E) return;
  const int row = __builtin_amdgcn_readfirstlane(ei[e]);
  const int col = __builtin_amdgcn_readfirstlane(ei[E + e]);
  const float w = m2p[0] * de2[row] * de3[col];
  const float2 v = *(const float2*)(xw + (size_t)row * DOUT + lane * 2);
  float* dst = h + (size_t)col * DOUT + lane * 2;
  unsafeAtomicAdd(dst,     w * v.x);
  unsafeAtomicAdd(dst + 1, w * v.y);
}

// ---------------------------------------------------------------------------
// Kernel 5: h = h_atomic + gso2*xw + 2b, then log_softmax over 64 classes.
// One wave32 per node, 2 classes/lane, shfl_xor wave reductions.
// ---------------------------------------------------------------------------
__global__ __launch_bounds__(256) void finalize_kernel(
    const float* __restrict__ xw, const float* __restrict__ gso2,
    const float* __restrict__ b, float* __restrict__ out, int N) {
  const int gid = blockIdx.x * 256 + threadIdx.x;
  const int node = gid >> 5;
  const int lane = gid & 31;
  if (node >= N) return;
  const float g = gso2[node];
  const size_t base = (size_t)node * DOUT + lane * 2;
  const float2 hv = *(const float2*)(out + base);
  const float2 xv = *(const float2*)(xw + base);
  const float t0 = hv.x + g * xv.x + 2.0f * b[lane * 2];
  const float t1 = hv.y + g * xv.y + 2.0f * b[lane * 2 + 1];

  float m = fmaxf(t0, t1);
  #pragma unroll
  for (int off = 16; off; off >>= 1) m = fmaxf(m, __shfl_xor(m, off, 32));
  float s = expf(t0 - m) + expf(t1 - m);
  #pragma unroll
  for (int off = 16; off; off >>= 1) s += __shfl_xor(s, off, 32);
  const float ls = logf(s);
  out[base]     = t0 - m - ls;
  out[base + 1] = t1 - m - ls;
}

// ---------------------------------------------------------------------------
extern "C" void kernel_launch(void* const* d_in, const int* in_sizes, int n_in,
                              void* d_out, int out_size, void* d_ws, size_t ws_size,
                              hipStream_t stream) {
  const float* x     = (const float*)d_in[0];
  const int*   ei    = (const int*)d_in[1];
  // d_in[2] = edge_index_id (arange self-loops) — folded analytically
  const float* diags = (const float*)d_in[3];
  const float* W     = (const float*)d_in[4];
  const float* b     = (const float*)d_in[5];
  const float* m1    = (const float*)d_in[6];
  const float* m2    = (const float*)d_in[7];
  const float* m3    = (const float*)d_in[8];
  const float* e1    = (const float*)d_in[9];
  const float* e2    = (const float*)d_in[10];
  const float* e3    = (const float*)d_in[11];

  const int N = in_sizes[3];
  const int E = in_sizes[1] / 2;

  float* ws   = (float*)d_ws;
  float* Wo   = ws;                       // 64*128
  float* de2  = Wo + DOUT * DIN;          // N
  float* de3  = de2 + N;                  // N
  float* gso2 = de3 + N;                  // N
  float* xw   = gso2 + N;                 // N*64
  float* out  = (float*)d_out;            // doubles as atomic accumulator

  const int total = N * DOUT;
  precompute_kernel<<<(total + 255) / 256, 256, 0, stream>>>(
      diags, out, de2, de3, gso2, m1, m2, m3, e1, e2, e3, N, total);

  bjorck_kernel<<<1, 256, 0, stream>>>(W, Wo);

  const int nRowTiles = (N + 15) / 16;
  gemm_xw_kernel<<<(nRowTiles + 7) / 8, 256, 0, stream>>>(x, Wo, xw, nRowTiles, N);

  const long long edgeThreads = (long long)E * 32;
  edge_scatter_kernel<<<(int)((edgeThreads + 255) / 256), 256, 0, stream>>>(
      ei, de2, de3, xw, m2, out, E);

  const long long nodeThreads = (long long)N * 32;
  finalize_kernel<<<(int)((nodeThreads + 255) / 256), 256, 0, stream>>>(
      xw, gso2, b, out, N);
}